// AggregationUnit_88407606821444
// MI455X (gfx1250) — compile-verified
//
#include <hip/hip_runtime.h>
#include <math.h>

// Problem constants (match reference)
#define B_ 2
#define C_ 32
#define H_ 64
#define W_ 64
#define P_ 3
#define P2_ 9
#define CD_ 288           // C_*P2_ descriptor length
#define DW_ 7
#define D2_ 49
#define K_ 4
#define NPIX 8192         // B_*H_*W_
#define HW_ 4096          // H_*W_
#define EPS_ 1e-12f
#define WIN_ 9            // staged window side: y-4..y+4 (disp +-3, patch +-1)
#define WINSZ (C_ * WIN_ * WIN_)   // 2592 floats = 10.4 KB

typedef float v2f __attribute__((ext_vector_type(2)));
typedef float v8f __attribute__((ext_vector_type(8)));

// ---------------------------------------------------------------------------
// Kernel 0: dnorm_map[b,y',x'] = max(||ftm1_u descriptor at (y',x')||, eps).
// The reference's dnorm[b,h,w,d] is position-only, so 8192 norms suffice
// (vs 8192*49 recomputations).
// ---------------------------------------------------------------------------
__global__ void __launch_bounds__(256)
dnorm_prepass(const float* __restrict__ feat_tm1, float* __restrict__ dnorm)
{
    const int n = blockIdx.x * blockDim.x + threadIdx.x;
    if (n >= NPIX) return;
    const int b = n >> 12, rem = n & 4095, y = rem >> 6, x = rem & 63;
    const float* f1 = feat_tm1 + b * (C_ * HW_);
    float ss = 0.f;
    for (int rr = 0; rr < P2_; ++rr) {
        const int yy = y + rr / 3 - 1;
        const int xx = x + rr % 3 - 1;
        if ((unsigned)yy < (unsigned)H_ && (unsigned)xx < (unsigned)W_) {
            const float* src = f1 + yy * W_ + xx;
            for (int c = 0; c < C_; ++c) {
                const float v = src[c * HW_];
                ss += v * v;
            }
        }
    }
    dnorm[n] = fmaxf(sqrtf(ss), EPS_);
}

// ---------------------------------------------------------------------------
// Kernel 1: wgt[o, n] = wproj_b[o] + sum_i wproj_w[o,i] * cat[i,n]
//   o in [0,288), i in [0,64), n = b*4096 + y*64 + x
// Pure fp32 GEMM on V_WMMA_F32_16X16X4_F32. One wave per 16x16 output tile,
// K=64 consumed as 16 chained k=4 WMMA steps. All lanes active (EXEC all-1s).
//
// fp32 A 16x4 layout (ISA 7.12.2): lanes 0-15 hold M=0..15, v0=K+0, v1=K+1;
// lanes 16-31 hold the same M rows with v0=K+2, v1=K+3. B (4x16) mirrors it:
// lane group selects the K pair, lane&15 selects N.
// C/D 16x16 f32: VGPR j -> M=j (lanes 0-15) / M=8+j (lanes 16-31), N=lane&15.
// ---------------------------------------------------------------------------
__global__ void __launch_bounds__(128)
wgt_gemm_wmma(const float* __restrict__ feat_t,
              const float* __restrict__ feat_tm1,
              const float* __restrict__ wproj_w,   // [288,64] row-major
              const float* __restrict__ wproj_b,   // [288]
              float* __restrict__ wgt)             // [288, 8192]
{
    const int wave = blockIdx.x * (blockDim.x >> 5) + (threadIdx.x >> 5);
    const int lane = threadIdx.x & 31;
    const int NT = NPIX / 16;            // 512 N tiles
    const int mt = wave % 18;            // 18 M tiles (288/16)
    const int nt = wave / 18;
    if (nt >= NT) return;                // wave-uniform exit

    const int m0 = mt * 16;
    const int n0 = nt * 16;
    const int hl = lane & 15;
    const int koff = (lane < 16) ? 0 : 2;

    const int n  = n0 + hl;              // this lane's pixel column
    const int bb = n >> 12;              // batch
    const int r  = n & 4095;             // y*64 + x
    const float* ft_base  = feat_t   + bb * (C_ * HW_) + r;
    const float* ftm_base = feat_tm1 + bb * (C_ * HW_) + r;

    v8f acc = {};
#pragma unroll
    for (int ks = 0; ks < 16; ++ks) {
        const int k0 = ks * 4 + koff;    // this lane's K pair base
        v2f a, bv;
        a.x = wproj_w[(m0 + hl) * 64 + k0];
        a.y = wproj_w[(m0 + hl) * 64 + k0 + 1];
        // cat[i, n]: i<32 -> feat_t channel i; i>=32 -> feat_tm1 channel i-32
        bv.x = (k0 < 32)     ? ft_base[k0 * HW_]        : ftm_base[(k0 - 32) * HW_];
        bv.y = (k0 + 1 < 32) ? ft_base[(k0 + 1) * HW_]  : ftm_base[(k0 - 31) * HW_];
        acc = __builtin_amdgcn_wmma_f32_16x16x4_f32(
            /*neg_a=*/false, a, /*neg_b=*/false, bv,
            /*c_mod=*/(short)0, acc, /*reuse_a=*/false, /*reuse_b=*/false);
    }

#pragma unroll
    for (int j = 0; j < 8; ++j) {
        const int row = m0 + ((lane < 16) ? j : (8 + j));
        wgt[row * NPIX + n] = acc[j] + wproj_b[row];
    }
}

// ---------------------------------------------------------------------------
// Kernel 2: fused correlation + top-4 + gather + aggregate + weighted sum.
// One 64-thread block (2 waves) per pixel. The 9x9x32 feat_tm1 window that
// feeds all 49 displacements AND the phase-4 gathers is staged once in LDS
// (zero-padded), converting ~49x-redundant VMEM traffic into DS reads.
// ---------------------------------------------------------------------------
__global__ void __launch_bounds__(64)
corr_topk_fused(const float* __restrict__ feat_t,
                const float* __restrict__ feat_tm1,
                const float* __restrict__ agg_w,    // [4]
                const float* __restrict__ agg_b,    // [1]
                const float* __restrict__ wgt,      // [288, 8192]
                const float* __restrict__ dnorm,    // [8192]
                float* __restrict__ out)            // [2,32,64,64]
{
    __shared__ float s_win[WINSZ];   // feat_tm1 window [c][wy][wx], wy=y-4+..
    __shared__ float s_ft[CD_];      // frame-t descriptor of this pixel
    __shared__ float s_red[64];      // norm reduction scratch
    __shared__ float s_sim[64];      // 49 similarities (+ pad)
    __shared__ int   s_idx[K_];
    __shared__ float s_ftn;

    const int tid = threadIdx.x;
    const int pix = blockIdx.x;                  // n = b*4096 + y*64 + x
    const int b   = pix >> 12;
    const int rem = pix & 4095;
    const int y   = rem >> 6;
    const int x   = rem & 63;
    const float* f1 = feat_tm1 + b * (C_ * HW_);

    // ---- Phase 0: stage zero-padded 9x9x32 feat_tm1 window into LDS
    for (int q = tid; q < WINSZ; q += 64) {
        const int c  = q / (WIN_ * WIN_);
        const int rw = q % (WIN_ * WIN_);
        const int yy = y - 4 + rw / WIN_;
        const int xx = x - 4 + rw % WIN_;
        float v = 0.f;
        if ((unsigned)yy < (unsigned)H_ && (unsigned)xx < (unsigned)W_)
            v = f1[c * HW_ + yy * W_ + xx];
        s_win[q] = v;
    }

    // ---- Phase 1: build ft_u descriptor (zero-padded 3x3 patch) + its norm
    float part = 0.f;
    for (int q = tid; q < CD_; q += 64) {
        const int c  = q / P2_;
        const int rr = q % P2_;
        const int yy = y + rr / 3 - 1;
        const int xx = x + rr % 3 - 1;
        float v = 0.f;
        if ((unsigned)yy < (unsigned)H_ && (unsigned)xx < (unsigned)W_)
            v = feat_t[(b * C_ + c) * HW_ + yy * W_ + xx];
        s_ft[q] = v;
        part += v * v;
    }
    s_red[tid] = part;
    __syncthreads();
    if (tid == 0) {
        float s = 0.f;
        for (int i = 0; i < 64; ++i) s += s_red[i];
        s_ftn = fmaxf(sqrtf(s), EPS_);
    }
    __syncthreads();
    const float ftn = s_ftn;

    // ---- Phase 2: one lane per displacement -> cosine similarity (LDS dots)
    if (tid < D2_) {
        const int di = tid / DW_, dj = tid % DW_;
        const int y2 = y + di - 3, x2 = x + dj - 3;
        float sim = 0.f;
        if ((unsigned)y2 < (unsigned)H_ && (unsigned)x2 < (unsigned)W_) {
            float dot = 0.f;
            for (int rr = 0; rr < P2_; ++rr) {
                const int w0 = (di + rr / 3) * WIN_ + (dj + rr % 3);
                for (int c = 0; c < C_; ++c)
                    dot += s_win[c * (WIN_ * WIN_) + w0] * s_ft[c * P2_ + rr];
            }
            sim = dot / (ftn * dnorm[b * HW_ + y2 * W_ + x2]);
        }
        s_sim[tid] = sim;      // off-frame displacement: sim = 0 (= 0/eps)
    } else {
        s_sim[tid] = -INFINITY;
    }
    __syncthreads();

    // ---- Phase 3: stable top-4 (matches lax.top_k: first max on ties)
    if (tid == 0) {
        for (int k = 0; k < K_; ++k) {
            float best = -INFINITY;
            int   bi   = 0;
            for (int d = 0; d < D2_; ++d) {
                if (s_sim[d] > best) { best = s_sim[d]; bi = d; }
            }
            s_idx[k] = bi;
            s_sim[bi] = -INFINITY;
        }
    }
    __syncthreads();

    // ---- Phase 4: one lane per channel -> gather (from LDS) + agg + wgt-sum
    if (tid < C_) {
        const int c = tid;
        const float ab = agg_b[0];
        float aw[K_];
        int   kdi[K_], kdj[K_];
        bool  kok[K_];
#pragma unroll
        for (int k = 0; k < K_; ++k) {
            aw[k] = agg_w[k];
            const int d = s_idx[k];
            kdi[k] = d / DW_;
            kdj[k] = d % DW_;
            const int y2 = y + kdi[k] - 3, x2 = x + kdj[k] - 3;
            kok[k] = (unsigned)y2 < (unsigned)H_ && (unsigned)x2 < (unsigned)W_;
        }
        const float* winc = s_win + c * (WIN_ * WIN_);
        float accum = 0.f;
#pragma unroll
        for (int p = 0; p < P2_; ++p) {
            const int pi = p / 3, pj = p % 3;
            const float w = wgt[(c * P2_ + p) * NPIX + pix];
            float a = ab;
#pragma unroll
            for (int k = 0; k < K_; ++k) {
                const float v = kok[k] ? winc[(kdi[k] + pi) * WIN_ + (kdj[k] + pj)] : 0.f;
                a += aw[k] * v;
            }
            accum += w * a;
        }
        out[(b * C_ + c) * HW_ + y * W_ + x] = accum;
    }
}

// ---------------------------------------------------------------------------
extern "C" void kernel_launch(void* const* d_in, const int* in_sizes, int n_in,
                              void* d_out, int out_size, void* d_ws, size_t ws_size,
                              hipStream_t stream)
{
    const float* feat_t   = (const float*)d_in[0];
    const float* feat_tm1 = (const float*)d_in[1];
    const float* agg_w    = (const float*)d_in[2];
    const float* agg_b    = (const float*)d_in[3];
    const float* wproj_w  = (const float*)d_in[4];
    const float* wproj_b  = (const float*)d_in[5];
    float* out   = (float*)d_out;
    float* wgt   = (float*)d_ws;                 // 288*8192 floats (9.44 MB)
    float* dnorm = wgt + (size_t)CD_ * NPIX;     // + 8192 floats
    (void)in_sizes; (void)n_in; (void)out_size; (void)ws_size;

    // Kernel 0: position-only descriptor norms of frame t-1 (8192 values).
    hipLaunchKernelGGL(dnorm_prepass, dim3(NPIX / 256), dim3(256), 0, stream,
                       feat_tm1, dnorm);

    // Kernel 1: 18*512 = 9216 WMMA output tiles, 4 waves per 128-thread block.
    const int waves  = 18 * (NPIX / 16);
    const int blocks = waves / 4;          // 2304
    hipLaunchKernelGGL(wgt_gemm_wmma, dim3(blocks), dim3(128), 0, stream,
                       feat_t, feat_tm1, wproj_w, wproj_b, wgt);

    // Kernel 2: one 64-thread block per pixel.
    hipLaunchKernelGGL(corr_topk_fused, dim3(NPIX), dim3(64), 0, stream,
                       feat_t, feat_tm1, agg_w, agg_b, wgt, dnorm, out);
}